// feature_encoding_72816875536993
// MI455X (gfx1250) — compile-verified
//
#include <hip/hip_runtime.h>

// Bilinear feature-grid sampling for MI455X (gfx1250).
//
// Replay-steady-state roofline: output (128 MB, write-once) is NT-stored so it
// rinses out of L2; feature grid (4 MB) + location stream (64 MB) together fit
// in the 192 MB L2 and are loaded with default RT policy so they stay resident
// across graph replays. Steady-state HBM traffic ~= 128 MB/launch -> ~5.5 us
// floor at 23.3 TB/s. WMMA is inapplicable: the blend is a batched 1x4 . 4x4
// product with a per-point B operand, nothing a 16x16 matrix pipe can share.

typedef __attribute__((ext_vector_type(2))) float v2f;
typedef __attribute__((ext_vector_type(4))) float v4f;

#define FG_H 512
#define FG_W 512

__device__ __forceinline__ v4f sample_one(const v4f* __restrict__ grid4, v2f p) {
    float x = p.x;
    float y = p.y;
    // x indexes dim0 (H), y indexes dim1 (W), matching the reference.
    float x0f = fminf(fmaxf(floorf(x), 0.0f), (float)(FG_H - 2));
    float y0f = fminf(fmaxf(floorf(y), 0.0f), (float)(FG_W - 2));
    float fx  = x - x0f;
    float fy  = y - y0f;

    int base = (int)x0f * FG_W + (int)y0f;

    // Four b128 gathers; the (y0,y1) pair is 32 contiguous bytes, so each
    // point usually touches two 64B cachelines. All L2-resident (4 MB grid).
    v4f f00 = grid4[base];
    v4f f01 = grid4[base + 1];
    v4f f10 = grid4[base + FG_W];
    v4f f11 = grid4[base + FG_W + 1];

    float gx = 1.0f - fx;
    float gy = 1.0f - fy;
    float w00 = gx * gy;
    float w01 = gx * fy;
    float w10 = fx * gy;
    float w11 = fx * fy;

    return f00 * w00 + f01 * w01 + f10 * w10 + f11 * w11;
}

__global__ __launch_bounds__(256) void bilinear_sample_gfx1250(
    const v4f* __restrict__ grid4,   // [FG_H*FG_W] 16-byte cells (4 features)
    const v2f* __restrict__ loc,     // [n] (x, y)
    v4f* __restrict__ out,           // [n]
    int n)
{
    const int tid    = blockIdx.x * blockDim.x + threadIdx.x;
    const int stride = gridDim.x * blockDim.x;

    int i = tid;

    // Main loop: two grid-stride points per iteration. Both location loads
    // issue back-to-back, then all 8 grid gathers — one loadcnt wait per
    // pair instead of per point, ~10 loads in flight per wave.
    for (; i + stride < n; i += 2 * stride) {
        const int j = i + stride;

        // Warm the next pair of location cachelines (speculative; OOB
        // translation failures are silently dropped).
        __builtin_prefetch((const void*)(loc + i + 2 * stride), /*rw=*/0, /*loc=*/3);

        // RT loads: locations are re-read every replay and (with the output
        // rinsed via NT stores) fit in L2 alongside the grid.
        v2f p0 = loc[i];
        v2f p1 = loc[j];

        v4f r0 = sample_one(grid4, p0);
        v4f r1 = sample_one(grid4, p1);

        // Write-once 128 MB stream: non-temporal, keep it out of L2.
        __builtin_nontemporal_store(r0, out + i);
        __builtin_nontemporal_store(r1, out + j);
    }

    // Tail (empty when n divides evenly across the launch).
    for (; i < n; i += stride) {
        v2f p = loc[i];
        v4f r = sample_one(grid4, p);
        __builtin_nontemporal_store(r, out + i);
    }
}

extern "C" void kernel_launch(void* const* d_in, const int* in_sizes, int n_in,
                              void* d_out, int out_size, void* d_ws, size_t ws_size,
                              hipStream_t stream) {
    (void)n_in; (void)out_size; (void)d_ws; (void)ws_size;

    const v4f* grid4 = (const v4f*)d_in[0];      // feature_grid: 512*512*4 f32
    const v2f* loc   = (const v2f*)d_in[1];      // location: [N,2] f32
    v4f*       out   = (v4f*)d_out;              // [N,4] f32

    const int n = in_sizes[1] / 2;               // N_POINTS

    const int threads = 256;                     // 8 wave32 waves per block
    // 4 points per thread => 2 paired iterations of the main loop.
    long long want_threads = ((long long)n + 3) / 4;
    int blocks = (int)((want_threads + threads - 1) / threads);
    if (blocks < 1) blocks = 1;

    bilinear_sample_gfx1250<<<blocks, threads, 0, stream>>>(grid4, loc, out, n);
}